// MRNet_77154792505971
// MI455X (gfx1250) — compile-verified
//
#include <hip/hip_runtime.h>
#include <hip/hip_bf16.h>
#include <math.h>

// ---------------------------------------------------------------------------
// Types for WMMA f32 16x16x4 (CDNA5 / gfx1250, wave32)
// ---------------------------------------------------------------------------
typedef float v2f __attribute__((ext_vector_type(2)));
typedef float v8f __attribute__((ext_vector_type(8)));

#define FEAT 128          // F_IN == HID == 128
#define KDIM 256          // concat(x, aggr) -> 2*128
#define ASTR 260          // LDS row stride (pad 4 floats -> avoid bank conflicts)

// Monotonic order-preserving float->int map: float compare == signed int
// compare on mapped values. map(-inf) = 0x807FFFFF (the fill pattern).
#define NEG_INF_MAPPED 0x807FFFFFu

__device__ __forceinline__ int mapOrd(float v) {
    int i = __float_as_int(v);
    return (i < 0) ? (i ^ 0x7FFFFFFF) : i;
}
__device__ __forceinline__ float unmapOrd(int i) {
    return __int_as_float((i < 0) ? (i ^ 0x7FFFFFFF) : i);
}

// ---------------------------------------------------------------------------
// Vectorized pattern fill (uint4 stores; counts here are multiples of 4)
// ---------------------------------------------------------------------------
__global__ void fill_pat4(uint32_t* __restrict__ p, uint32_t pat, int count4) {
    int i = blockIdx.x * blockDim.x + threadIdx.x;
    if (i < count4) {
        uint4 v = {pat, pat, pat, pat};
        *(uint4*)&p[(size_t)i * 4] = v;
    }
}

// ---------------------------------------------------------------------------
// Edge message + scatter-max.  One wave per undirected edge; each lane owns a
// float4 chunk of the 128 features.  Both directions emitted (to_undirected).
// Scatter-max is a single branchless global_atomic_max_i32 on mapped values.
// ---------------------------------------------------------------------------
__global__ void edge_scatter(const float* __restrict__ feat,
                             const int* __restrict__ src,
                             const int* __restrict__ dst,
                             int* __restrict__ aggrOrd, int E) {
    int wave = threadIdx.x >> 5;
    int lane = threadIdx.x & 31;
    int e = blockIdx.x * 8 + wave;
    if (e >= E) return;
    int s = src[e];
    int d = dst[e];
    const float4 xs = *(const float4*)&feat[(size_t)s * FEAT + lane * 4];
    const float4 xd = *(const float4*)&feat[(size_t)d * FEAT + lane * 4];
    float df[4] = {xs.x - xd.x, xs.y - xd.y, xs.z - xd.z, xs.w - xd.w};
    int* ad = &aggrOrd[(size_t)d * FEAT + lane * 4];
    int* as = &aggrOrd[(size_t)s * FEAT + lane * 4];
#pragma unroll
    for (int c = 0; c < 4; ++c) {
        atomicMax(ad + c, mapOrd(df[c]));
        atomicMax(as + c, mapOrd(-df[c]));
    }
}

// ---------------------------------------------------------------------------
// Fused  out = relu( [feat | clean(aggr)] @ W + b )  using V_WMMA_F32_16X16X4.
//  - block: 256 threads = 8 waves; block owns a 16-row strip of M
//  - wave w owns output columns [16w, 16w+16)  (N = 128 -> 8 waves)
//  - A tile (16 x 256) staged in LDS once per block; aggr unmapped from the
//    order-preserving int domain and -inf (empty segment) zeroed at stage time
//  - B read from global (128 KB weight matrix stays L2-resident)
// Layouts per CDNA5 ISA 7.12.2:
//  A 16x4 : lanes 0-15 K={0,1}, lanes 16-31 K={2,3}; VGPR j -> K = 2*half + j
//  B 4x16 : mirrored; C 16x16: VGPR j -> row j (+8 upper half), N = lane&15
// ---------------------------------------------------------------------------
__global__ void mrconv_gemm(const float* __restrict__ feat,
                            const int* __restrict__ aggrOrd,
                            const float* __restrict__ W,
                            const float* __restrict__ bias,
                            float* __restrict__ out, int N) {
    __shared__ float sA[16 * ASTR];

    const int m0   = blockIdx.x * 16;
    const int wave = threadIdx.x >> 5;
    const int lane = threadIdx.x & 31;
    const int half = lane >> 4;     // 0: lanes 0-15, 1: lanes 16-31
    const int l16  = lane & 15;
    const int n0   = wave * 16;

    // Cooperative stage of A = [feat | aggr(unmapped, cleaned)]
    for (int i = 0; i < 16; ++i) {
        const int k = threadIdx.x;          // 0..255
        const int m = m0 + i;
        float v = 0.0f;
        if (m < N) {                        // block-uniform condition
            if (k < FEAT) {
                v = feat[(size_t)m * FEAT + k];
            } else {
                v = unmapOrd(aggrOrd[(size_t)m * FEAT + (k - FEAT)]);
                if (!isfinite(v)) v = 0.0f; // empty-segment max -> 0 (PyG)
            }
        }
        sA[i * ASTR + k] = v;
    }
    __syncthreads();

    v8f acc = {};   // C accumulator (16x16 f32 tile, 8 VGPRs)
#pragma unroll 4
    for (int k0 = 0; k0 < KDIM; k0 += 4) {
        const int kk = k0 + 2 * half;
        v2f a;
        a.x = sA[l16 * ASTR + kk];
        a.y = sA[l16 * ASTR + kk + 1];
        v2f b;
        b.x = W[(size_t)kk * FEAT + n0 + l16];
        b.y = W[(size_t)(kk + 1) * FEAT + n0 + l16];
        acc = __builtin_amdgcn_wmma_f32_16x16x4_f32(
            /*neg_a=*/false, a, /*neg_b=*/false, b,
            /*c_mod=*/(short)0, acc, /*reuse_a=*/false, /*reuse_b=*/false);
    }

    const float bv = bias[n0 + l16];
    if (m0 + 16 <= N) {
        // Full tile (always true for N % 16 == 0): unguarded, no exec churn
#pragma unroll
        for (int j = 0; j < 8; ++j) {
            const int m = m0 + j + half * 8;
            float v = acc[j] + bv;
            out[(size_t)m * FEAT + n0 + l16] = v > 0.0f ? v : 0.0f;
        }
    } else {
#pragma unroll
        for (int j = 0; j < 8; ++j) {
            const int m = m0 + j + half * 8;
            if (m < N) {
                float v = acc[j] + bv;
                out[(size_t)m * FEAT + n0 + l16] = v > 0.0f ? v : 0.0f;
            }
        }
    }
}

// ---------------------------------------------------------------------------
// global_max_pool: h >= 0 (post-ReLU) and pool buffer pre-zeroed, so plain
// signed-int atomicMax is exact (and matches empty-segment -> 0).
// ---------------------------------------------------------------------------
__global__ void pool_max(const float* __restrict__ h,
                         const int* __restrict__ batch,
                         float* __restrict__ g, int N) {
    int idx = blockIdx.x * blockDim.x + threadIdx.x;
    if (idx >= N * FEAT) return;
    int n = idx >> 7;       // / 128
    int f = idx & 127;
    int b = batch[n];
    atomicMax((int*)&g[(size_t)b * FEAT + f], __float_as_int(h[idx]));
}

// ---------------------------------------------------------------------------
// Head MLP + log_softmax: one block per graph (64 graphs, trivial work).
// ---------------------------------------------------------------------------
__global__ void mlp_head(const float* __restrict__ g,
                         const float* __restrict__ Wf1, const float* __restrict__ bf1,
                         const float* __restrict__ Wf2, const float* __restrict__ bf2,
                         float* __restrict__ out) {
    __shared__ float hid[64];
    __shared__ float logit[40];
    const int gr = blockIdx.x;
    const int t  = threadIdx.x;      // 64 threads

    float a1 = bf1[t];
    const float* grow = &g[(size_t)gr * FEAT];
    for (int k = 0; k < FEAT; ++k) a1 = fmaf(grow[k], Wf1[k * 64 + t], a1);
    hid[t] = fmaxf(a1, 0.0f);
    __syncthreads();

    if (t < 40) {
        float a2 = bf2[t];
        for (int k = 0; k < 64; ++k) a2 = fmaf(hid[k], Wf2[k * 40 + t], a2);
        logit[t] = a2;
    }
    __syncthreads();

    if (t == 0) {
        float mx = -INFINITY;
        for (int c = 0; c < 40; ++c) mx = fmaxf(mx, logit[c]);
        float se = 0.0f;
        for (int c = 0; c < 40; ++c) se += expf(logit[c] - mx);
        float lse = mx + logf(se);
        for (int c = 0; c < 40; ++c) out[(size_t)gr * 40 + c] = logit[c] - lse;
    }
}

// ---------------------------------------------------------------------------
// Launch: 3 x (fill mapped(-inf), edge scatter-max, WMMA GEMM+ReLU),
// then pool + head.
// ---------------------------------------------------------------------------
extern "C" void kernel_launch(void* const* d_in, const int* in_sizes, int n_in,
                              void* d_out, int out_size, void* d_ws, size_t ws_size,
                              hipStream_t stream) {
    const float* x     = (const float*)d_in[0];
    const int*   ei    = (const int*)d_in[1];
    const int*   batch = (const int*)d_in[2];
    const float* W1 = (const float*)d_in[3];  const float* b1 = (const float*)d_in[4];
    const float* W2 = (const float*)d_in[5];  const float* b2 = (const float*)d_in[6];
    const float* W3 = (const float*)d_in[7];  const float* b3 = (const float*)d_in[8];
    const float* Wf1 = (const float*)d_in[9];  const float* bf1 = (const float*)d_in[10];
    const float* Wf2 = (const float*)d_in[11]; const float* bf2 = (const float*)d_in[12];
    float* out = (float*)d_out;

    const int N = in_sizes[0] / FEAT;   // 50000
    const int E = in_sizes[1] / 2;      // 800000
    const int G = out_size / 40;        // 64
    const int NF = N * FEAT;

    const int* src = ei;
    const int* dst = ei + E;

    // Workspace carve-up
    int*   aggr = (int*)d_ws;                       // mapped-order ints
    float* hA   = (float*)(aggr + (size_t)NF);
    float* hB   = hA + (size_t)NF;
    float* g    = hB + (size_t)NF;

    const int fillBlocks = (NF / 4 + 255) / 256;
    const int edgeBlocks = (E + 7) / 8;       // 8 waves/block, wave per edge
    const int gemmBlocks = (N + 15) / 16;     // 16-row strips (50000/16 exact)
    const int poolBlocks = (NF + 255) / 256;

    // ---- Layer 1: x -> hA
    fill_pat4<<<fillBlocks, 256, 0, stream>>>((uint32_t*)aggr, NEG_INF_MAPPED, NF / 4);
    edge_scatter<<<edgeBlocks, 256, 0, stream>>>(x, src, dst, aggr, E);
    mrconv_gemm<<<gemmBlocks, 256, 0, stream>>>(x, aggr, W1, b1, hA, N);

    // ---- Layer 2: hA -> hB
    fill_pat4<<<fillBlocks, 256, 0, stream>>>((uint32_t*)aggr, NEG_INF_MAPPED, NF / 4);
    edge_scatter<<<edgeBlocks, 256, 0, stream>>>(hA, src, dst, aggr, E);
    mrconv_gemm<<<gemmBlocks, 256, 0, stream>>>(hA, aggr, W2, b2, hB, N);

    // ---- Layer 3: hB -> hA
    fill_pat4<<<fillBlocks, 256, 0, stream>>>((uint32_t*)aggr, NEG_INF_MAPPED, NF / 4);
    edge_scatter<<<edgeBlocks, 256, 0, stream>>>(hB, src, dst, aggr, E);
    mrconv_gemm<<<gemmBlocks, 256, 0, stream>>>(hB, aggr, W3, b3, hA, N);

    // ---- Global max pool (post-ReLU h >= 0, init 0 + int max is exact)
    fill_pat4<<<(G * FEAT / 4 + 255) / 256, 256, 0, stream>>>((uint32_t*)g, 0u, G * FEAT / 4);
    pool_max<<<poolBlocks, 256, 0, stream>>>(hA, batch, g, N);

    // ---- Head MLP + log_softmax
    mlp_head<<<G, 64, 0, stream>>>(g, Wf1, bf1, Wf2, bf2, out);
}